// ScaledDotProductAttention_20426864460520
// MI455X (gfx1250) — compile-verified
//
#include <hip/hip_runtime.h>
#include <hip/hip_bf16.h>
#include <cstdint>

typedef __attribute__((ext_vector_type(2))) float v2f;
typedef __attribute__((ext_vector_type(8))) float v8f;

#define B_    8
#define H_    16
#define LQ_   1024
#define LK_   1024
#define DK_   64
#define QTILE 32
#define SPITCH 1032   // padded LDS row stride (floats): 1032 % 64 == 8 -> conflict-free softmax

extern __shared__ float s_scores[];   // [QTILE][SPITCH] dynamic LDS (129 KB of 320 KB WGP LDS)

__global__ __launch_bounds__(256)
void attn_f32wmma_kernel(const float* __restrict__ q,
                         const float* __restrict__ k,
                         const float* __restrict__ v,
                         const float* __restrict__ prev,
                         const unsigned char* __restrict__ mask,
                         const float* __restrict__ scale_p,
                         float* __restrict__ out,
                         float* __restrict__ wts,
                         float* __restrict__ scr)
{
    __shared__ float s_red[QTILE][8];

    const int tid  = threadIdx.x;
    const int lane = tid & 31;
    const int w    = tid >> 5;       // wave id 0..7
    const int lo   = lane & 15;      // 0..15
    const int hi   = lane >> 4;      // 0..1 (fp32 WMMA half-wave K split)
    const int qt   = blockIdx.x;     // 0..31  (LQ/QTILE)
    const int bh   = blockIdx.y;     // 0..127 (B*H)
    const int b    = bh >> 4;        // bh / H
    const int qbase = qt * QTILE;
    const float scale = scale_p[0];

    const float* qp = q    + (size_t)bh * LQ_ * DK_;
    const float* kp = k    + (size_t)bh * DK_ * LK_;
    const float* vp = v    + (size_t)bh * LK_ * DK_;
    const float* pp = prev + (size_t)bh * LQ_ * LK_;
    const unsigned char* mp = mask + (size_t)b * LK_;
    float* op = out + (size_t)bh * LQ_ * DK_;
    float* wp = wts + (size_t)bh * LQ_ * LK_;
    float* sp = scr + (size_t)bh * LQ_ * LK_;

    const float neg_inf = -__builtin_huge_valf();

    // ---------------- Phase A: S = scale * (Q @ K) + prev, mask -> -inf ----------------
    {
        const int qsub = w >> 2;                 // 0..1 : 16-row q subtile
        const int ctg  = w & 3;                  // 0..3 : 256-col stripe
        const int qrow = qbase + qsub * 16 + lo;

        // A fragments for full K=64 (16 steps of 4): reused across all 16 column tiles.
        // fp32 A 16x4 layout: lanes 0-15 hold K={0,1}, lanes 16-31 hold K={2,3}.
        v2f a[16];
        #pragma unroll
        for (int kk = 0; kk < 16; ++kk) {
            a[kk] = *(const v2f*)(qp + (size_t)qrow * DK_ + 4 * kk + 2 * hi);
        }

        #pragma unroll 1
        for (int j = 0; j < 16; ++j) {
            const int colbase = (ctg * 16 + j) * 16;
            const int col = colbase + lo;
            v8f acc = {0.f, 0.f, 0.f, 0.f, 0.f, 0.f, 0.f, 0.f};
            #pragma unroll
            for (int kk = 0; kk < 16; ++kk) {
                const int krow = 4 * kk + 2 * hi;
                v2f bf;
                bf.x = kp[(size_t)krow * LK_ + col];        // keys stored [DK, LK] == B layout
                bf.y = kp[(size_t)(krow + 1) * LK_ + col];
                acc = __builtin_amdgcn_wmma_f32_16x16x4_f32(
                        false, a[kk], false, bf, (short)0, acc, false, false);
            }
            const bool masked = (mp[col] != 0);
            #pragma unroll
            for (int r = 0; r < 8; ++r) {
                const int row = qsub * 16 + r + 8 * hi;     // row within 32-row tile
                float s = acc[r] * scale + pp[(size_t)(qbase + row) * LK_ + col];
                if (masked) s = neg_inf;
                sp[(size_t)(qbase + row) * LK_ + col] = s;  // attn_scores output
                s_scores[row * SPITCH + col] = s;           // keep for softmax
            }
        }
    }
    __syncthreads();

    // ---------------- Phase B: row softmax (32 rows x 1024, 8 threads/row) ----------------
    {
        const int r   = tid >> 3;   // row 0..31
        const int seg = tid & 7;    // 8-way column interleave

        float m = neg_inf;
        #pragma unroll 8
        for (int i = 0; i < 128; ++i)
            m = fmaxf(m, s_scores[r * SPITCH + seg + 8 * i]);
        s_red[r][seg] = m;
        __syncthreads();
        m = s_red[r][0];
        #pragma unroll
        for (int t = 1; t < 8; ++t) m = fmaxf(m, s_red[r][t]);
        __syncthreads();   // s_red reuse

        float sum = 0.f;
        #pragma unroll 8
        for (int i = 0; i < 128; ++i) {
            const int c = seg + 8 * i;
            const float e = __expf(s_scores[r * SPITCH + c] - m);
            s_scores[r * SPITCH + c] = e;
            sum += e;
        }
        s_red[r][seg] = sum;
        __syncthreads();
        sum = 0.f;
        #pragma unroll
        for (int t = 0; t < 8; ++t) sum += s_red[r][t];
        const float inv = 1.0f / sum;

        #pragma unroll 8
        for (int i = 0; i < 128; ++i) {
            const int c = seg + 8 * i;
            const float wv = s_scores[r * SPITCH + c] * inv;
            s_scores[r * SPITCH + c] = wv;                  // keep for P@V
            wp[(size_t)(qbase + r) * LK_ + c] = wv;         // attn_weights output
        }
    }
    __syncthreads();

    // ---------------- Phase C: O = P @ V  (K = 1024 in 256 fp32 WMMAs) ----------------
    {
        const int qsub = w >> 2;    // 0..1
        const int dkt  = w & 3;     // 0..3 : 16-wide DK tile
        v8f acc = {0.f, 0.f, 0.f, 0.f, 0.f, 0.f, 0.f, 0.f};
        #pragma unroll 4
        for (int kt = 0; kt < 256; ++kt) {
            const int k0 = 4 * kt + 2 * hi;
            const v2f a2 = *(const v2f*)&s_scores[(qsub * 16 + lo) * SPITCH + k0];
            v2f b2;
            b2.x = vp[(size_t)k0 * DK_ + dkt * 16 + lo];    // values stored [LK, DK] == B layout
            b2.y = vp[(size_t)(k0 + 1) * DK_ + dkt * 16 + lo];
            acc = __builtin_amdgcn_wmma_f32_16x16x4_f32(
                    false, a2, false, b2, (short)0, acc, false, false);
        }
        #pragma unroll
        for (int r = 0; r < 8; ++r) {
            const int row = qbase + qsub * 16 + r + 8 * hi;
            op[(size_t)row * DK_ + dkt * 16 + lo] = acc[r];
        }
    }
}

extern "C" void kernel_launch(void* const* d_in, const int* in_sizes, int n_in,
                              void* d_out, int out_size, void* d_ws, size_t ws_size,
                              hipStream_t stream)
{
    (void)in_sizes; (void)n_in; (void)d_ws; (void)ws_size; (void)out_size;

    const float* q    = (const float*)d_in[0];
    const float* k    = (const float*)d_in[1];
    const float* v    = (const float*)d_in[2];
    const float* prev = (const float*)d_in[3];
    const unsigned char* mask = (const unsigned char*)d_in[4];
    const float* scal = (const float*)d_in[5];

    float* out = (float*)d_out;                                   // [B,H,LQ,DK]
    float* wts = out + (size_t)B_ * H_ * LQ_ * DK_;               // [B,H,LQ,LK]
    float* scr = wts + (size_t)B_ * H_ * LQ_ * LK_;               // [B,H,LQ,LK]

    dim3 grid(LQ_ / QTILE, B_ * H_);
    size_t shmem = (size_t)QTILE * SPITCH * sizeof(float);        // 132096 B dynamic LDS
    attn_f32wmma_kernel<<<grid, dim3(256), shmem, stream>>>(
        q, k, v, prev, mask, scal, out, wts, scr);
}